// NPCLoss_47648367182235
// MI455X (gfx1250) — compile-verified
//
#include <hip/hip_runtime.h>
#include <math.h>

#define BDIM 256
#define TILE_FLOATS (BDIM * 4)   // 1024 floats = 4 KB per tile
#define NSORT 4096               // pow2 >= B for bitonic sort

// ---------------------------------------------------------------------------
// Kernel 1: per-row stats. One block per row, 256 threads (8 wave32).
// Streams the row through double-buffered LDS using CDNA5 async loads with
// non-temporal hints (823 MB single-use stream; don't pollute L2/WGP$).
// ---------------------------------------------------------------------------
__global__ __launch_bounds__(BDIM) void npc_row_stats(
    const float* __restrict__ outp, const int* __restrict__ tgt,
    float* __restrict__ l_arr, unsigned* __restrict__ n_neg, int B, int C)
{
    __shared__ float buf[2][TILE_FLOATS];
    __shared__ float red_m[8], red_s[8], red_mx[8], red_c[8];

    const int row = blockIdx.x;
    const int tid = threadIdx.x;
    const int target = tgt[row];
    const float* rowp = outp + (size_t)row * (size_t)C;

    const int fullTiles = C / TILE_FLOATS;

    float m = -INFINITY;     // running max for logsumexp
    float s = 0.0f;          // running sum of exp(x - m)
    float mnt = -INFINITY;   // max over non-target columns
    float corr = -INFINITY;  // true-class logit (only one lane sees it)

    // Prologue: issue async load of tile 0 into buf[0] (one VMEM op per wave,
    // each lane writes its own 16B slice of LDS). Tracked by ASYNCcnt.
    {
        unsigned loff = (unsigned)(size_t)(&buf[0][tid * 4]);
        unsigned long long ga =
            (unsigned long long)(size_t)(rowp + (size_t)tid * 4);
        asm volatile("global_load_async_to_lds_b128 %0, %1, off th:TH_LOAD_NT"
                     :: "v"(loff), "v"(ga) : "memory");
    }

    for (int t = 0; t < fullTiles; ++t) {
        if (t + 1 < fullTiles) {
            // issue next tile, then wait until only it is outstanding
            unsigned loff = (unsigned)(size_t)(&buf[(t + 1) & 1][tid * 4]);
            unsigned long long ga = (unsigned long long)(size_t)(
                rowp + (size_t)(t + 1) * TILE_FLOATS + (size_t)tid * 4);
            asm volatile("global_load_async_to_lds_b128 %0, %1, off th:TH_LOAD_NT"
                         :: "v"(loff), "v"(ga) : "memory");
            asm volatile("s_wait_asynccnt 1" ::: "memory");
        } else {
            asm volatile("s_wait_asynccnt 0" ::: "memory");
        }

        const float4 v = *(const float4*)(&buf[t & 1][tid * 4]);
        const int colbase = t * TILE_FLOATS + tid * 4;
        const float x0 = v.x, x1 = v.y, x2 = v.z, x3 = v.w;

        // group-max online logsumexp: 1 rescale + 4 exps per 4 elements
        const float gm = fmaxf(fmaxf(x0, x1), fmaxf(x2, x3));
        const float nm = fmaxf(m, gm);
        const float e0 = __expf(x0 - nm);
        const float e1 = __expf(x1 - nm);
        const float e2 = __expf(x2 - nm);
        const float e3 = __expf(x3 - nm);
        s = s * __expf(m - nm) + ((e0 + e1) + (e2 + e3));
        m = nm;

        // masked max (exclude target) + true-logit capture, full-rate VALU
        const float x[4] = {x0, x1, x2, x3};
#pragma unroll
        for (int k = 0; k < 4; ++k) {
            const int col = colbase + k;
            if (col == target) corr = x[k];
            else               mnt = fmaxf(mnt, x[k]);
        }
    }

    // Tail: C % TILE_FLOATS elements via guarded non-temporal loads
    for (int col = fullTiles * TILE_FLOATS + tid; col < C; col += BDIM) {
        const float xv = __builtin_nontemporal_load(&rowp[col]);
        const float nm = fmaxf(m, xv);
        s = s * __expf(m - nm) + __expf(xv - nm);
        m = nm;
        if (col == target) corr = xv;
        else               mnt = fmaxf(mnt, xv);
    }

    // wave32 reduction
#pragma unroll
    for (int off = 16; off > 0; off >>= 1) {
        float m2 = __shfl_xor(m, off, 32);
        float s2 = __shfl_xor(s, off, 32);
        float nm = fmaxf(m, m2);
        s = s * __expf(m - nm) + s2 * __expf(m2 - nm);
        m = nm;
        mnt  = fmaxf(mnt,  __shfl_xor(mnt,  off, 32));
        corr = fmaxf(corr, __shfl_xor(corr, off, 32));
    }
    const int wave = tid >> 5;
    if ((tid & 31) == 0) {
        red_m[wave] = m; red_s[wave] = s; red_mx[wave] = mnt; red_c[wave] = corr;
    }
    __syncthreads();

    if (tid == 0) {
        float M = red_m[0], S = red_s[0], MX = red_mx[0], CR = red_c[0];
#pragma unroll
        for (int w = 1; w < 8; ++w) {
            float nm = fmaxf(M, red_m[w]);
            S = S * __expf(M - nm) + red_s[w] * __expf(red_m[w] - nm);
            M = nm;
            MX = fmaxf(MX, red_mx[w]);
            CR = fmaxf(CR, red_c[w]);
        }
        const float lse = M + __logf(S);
        const float margin = CR - MX;
        float l = (margin >= 0.0f) ? (1.0f - margin) : (1.0f - CR + lse);
        l = fmaxf(l, 0.0f);
        l_arr[row] = l;
        if (margin < 0.0f) atomicAdd(n_neg, 1u);
    }
}

// ---------------------------------------------------------------------------
// Kernel 2: sort + cumsum + thresholded selection. One block of 1024 threads.
// ---------------------------------------------------------------------------
__global__ __launch_bounds__(1024) void npc_select(
    const float* __restrict__ l_arr, const unsigned* __restrict__ n_neg,
    float* __restrict__ outv, int B)
{
    __shared__ float a[NSORT];
    __shared__ float part[1024];
    __shared__ float redA[32], redB[32];

    const int tid = threadIdx.x;

    for (int i = tid; i < NSORT; i += 1024)
        a[i] = (i < B) ? l_arr[i] : INFINITY;
    __syncthreads();

    // Bitonic sort ascending (pairs within a (k,j) step are disjoint)
    for (int k = 2; k <= NSORT; k <<= 1) {
        for (int j = k >> 1; j > 0; j >>= 1) {
            for (int i = tid; i < NSORT; i += 1024) {
                const int ixj = i ^ j;
                if (ixj > i) {
                    const float x = a[i], y = a[ixj];
                    const bool up = ((i & k) == 0);
                    if ((x > y) == up) { a[i] = y; a[ixj] = x; }
                }
            }
            __syncthreads();
        }
    }

    // Blocked inclusive scan: 4 elems/thread + Hillis-Steele over partials
    const int base4 = tid * 4;
    const float x0 = a[base4 + 0], x1 = a[base4 + 1],
                x2 = a[base4 + 2], x3 = a[base4 + 3];
    const float s0 = x0, s1 = s0 + x1, s2 = s1 + x2, s3 = s2 + x3;
    part[tid] = s3;
    __syncthreads();
    for (int off = 1; off < 1024; off <<= 1) {
        const float v   = part[tid];
        const float add = (tid >= off) ? part[tid - off] : 0.0f;
        __syncthreads();
        part[tid] = v + add;
        __syncthreads();
    }
    const float excl = (tid > 0) ? part[tid - 1] : 0.0f;

    // threshold = floor((1-eps)^2 * B + (1-eps) * n_neg), eps = 0.3
    const float neg = (float)(*n_neg);
    const float thr = floorf(0.49f * (float)B + 0.7f * neg);

    const float c[4] = {excl + s0, excl + s1, excl + s2, excl + s3};
    const float xv[4] = {x0, x1, x2, x3};
    float ksum = 0.0f, kcnt = 0.0f;
#pragma unroll
    for (int kk = 0; kk < 4; ++kk) {
        const int i = base4 + kk;
        if (i < B && (c[kk] <= thr + 1.0f - (float)i)) {
            ksum += xv[kk];
            kcnt += 1.0f;
        }
    }

    // reduce ksum / kcnt across 1024 threads
#pragma unroll
    for (int off = 16; off > 0; off >>= 1) {
        ksum += __shfl_xor(ksum, off, 32);
        kcnt += __shfl_xor(kcnt, off, 32);
    }
    if ((tid & 31) == 0) { redA[tid >> 5] = ksum; redB[tid >> 5] = kcnt; }
    __syncthreads();
    if (tid < 32) {
        float ka = redA[tid], kb = redB[tid];
#pragma unroll
        for (int off = 16; off > 0; off >>= 1) {
            ka += __shfl_xor(ka, off, 32);
            kb += __shfl_xor(kb, off, 32);
        }
        if (tid == 0) {
            const float npcl_1 = ka;
            const float npcl_2 = thr - kb;
            outv[0] = (npcl_1 < npcl_2) ? npcl_2 : npcl_1;
        }
    }
}

// ---------------------------------------------------------------------------
extern "C" void kernel_launch(void* const* d_in, const int* in_sizes, int n_in,
                              void* d_out, int out_size, void* d_ws, size_t ws_size,
                              hipStream_t stream) {
    const float* outp = (const float*)d_in[0];
    const int*   tgt  = (const int*)d_in[1];
    float*       res  = (float*)d_out;

    const int B = in_sizes[1];
    const int C = in_sizes[0] / B;

    unsigned* n_neg = (unsigned*)d_ws;
    float*    l_arr = (float*)((char*)d_ws + 256);

    hipMemsetAsync(d_ws, 0, 256, stream);
    npc_row_stats<<<B, BDIM, 0, stream>>>(outp, tgt, l_arr, n_neg, B, C);
    npc_select<<<1, 1024, 0, stream>>>(l_arr, n_neg, res, B);
}